// GGNN_43319040147495
// MI455X (gfx1250) — compile-verified
//
#include <hip/hip_runtime.h>
#include <hip/hip_bf16.h>
#include <math.h>

// GGNN: N=18 nodes, C=4 classes, A=14 alarms, H=20 hidden, O=8 out, T=5 steps.
// Single-workgroup, single-wave latency-bound kernel. All matmuls mapped to
// V_WMMA_F32_16X16X4_F32 (native fp32 WMMA on CDNA5), data staged in LDS with
// zero padding to 16-multiples so padding is invariant through the recurrence.

typedef __attribute__((ext_vector_type(2))) float v2f;
typedef __attribute__((ext_vector_type(8))) float v8f;

#define ACT_NONE 0
#define ACT_SIG  1
#define ACT_TANH 2

__device__ __forceinline__ float sigf(float v) {
    return 1.0f / (1.0f + __expf(-v));
}

// Accumulate one 16x16 output tile: c += A[row0:row0+16, 0:K] * B[0:K, col0:col0+16]
// K must be a multiple of 4. If atrans, A is read transposed (element (m,k) = A[k*lda+m]).
// VGPR layouts per CDNA5 ISA 7.12.2:
//   A 16x4:  lanes 0-15 hold (M=lane, K=0/1), lanes 16-31 hold (M=lane-16, K=2/3)
//   B 4x16:  v0 = rows K=0 (lanes 0-15) / K=2 (lanes 16-31); v1 = K=1 / K=3
//   C 16x16: v[j] = row M=j (lanes 0-15) / M=j+8 (lanes 16-31), N=lane&15
__device__ __forceinline__ v8f wmma_acc(v8f c,
                                        const float* __restrict__ A, int lda, int row0,
                                        const float* __restrict__ B, int ldb, int col0,
                                        int K, bool atrans) {
    const int lane  = threadIdx.x & 31;
    const int half2 = (lane >> 4) << 1;   // 0 or 2
    const int l     = lane & 15;
    for (int k = 0; k < K; k += 4) {
        const int kk = k + half2;
        v2f av, bv;
        if (!atrans) {
            av.x = A[(row0 + l) * lda + kk];
            av.y = A[(row0 + l) * lda + kk + 1];
        } else {
            av.x = A[kk * lda + row0 + l];
            av.y = A[(kk + 1) * lda + row0 + l];
        }
        bv.x = B[kk * ldb + col0 + l];
        bv.y = B[(kk + 1) * ldb + col0 + l];
        // (neg_a, A, neg_b, B, c_mod, C, reuse_a, reuse_b)
        c = __builtin_amdgcn_wmma_f32_16x16x4_f32(false, av, false, bv,
                                                  (short)0, c, false, false);
    }
    return c;
}

__device__ __forceinline__ void store_tile(float* __restrict__ D, int ldd,
                                           int row0, int col0, v8f c,
                                           const float* bias, int act) {
    const int lane = threadIdx.x & 31;
    const int half = lane >> 4;
    const int l    = lane & 15;
    const float b  = bias ? bias[col0 + l] : 0.0f;
#pragma unroll
    for (int j = 0; j < 8; ++j) {
        float v = c[j] + b;
        if (act == ACT_SIG)  v = sigf(v);
        if (act == ACT_TANH) v = tanhf(v);
        D[(row0 + j + 8 * half) * ldd + col0 + l] = v;
    }
}

__global__ void __launch_bounds__(32)
ggnn_kernel(const float* __restrict__ x,   const float* __restrict__ adj,
            const float* __restrict__ Wz,  const float* __restrict__ bz,
            const float* __restrict__ Uz,  const float* __restrict__ buz,
            const float* __restrict__ Wr,  const float* __restrict__ br,
            const float* __restrict__ Ur,  const float* __restrict__ bur,
            const float* __restrict__ Ww,  const float* __restrict__ bw,
            const float* __restrict__ Uw,  const float* __restrict__ bu,
            const float* __restrict__ Wo,  const float* __restrict__ bo,
            const float* __restrict__ Wx,  const float* __restrict__ bx,
            const float* __restrict__ Wy,  const float* __restrict__ by,
            const float* __restrict__ w2,  const float* __restrict__ b2,
            float* __restrict__ out) {
    // Padded dims: N 18->32, H 20->32, 2H=40 (already mult of 4), O 8->16.
    __shared__ float s_adj[32 * 32];   // in_mat (padded adjacency)
    __shared__ float s_h[32 * 32];     // hidden state
    __shared__ float s_x[32 * 32];     // initial x (for final concat)
    __shared__ float s_ain[32 * 32];   // in_mat @ h
    __shared__ float s_aout[32 * 32];  // in_mat^T @ h
    __shared__ float s_z[32 * 32];
    __shared__ float s_r[32 * 32];     // r, then r*h in place
    __shared__ float s_hg[32 * 32];
    __shared__ float s_WzT[40 * 32];   // Wz^T  [k][n]
    __shared__ float s_WrT[40 * 32];
    __shared__ float s_WwT[40 * 32];
    __shared__ float s_UzT[20 * 32];
    __shared__ float s_UrT[20 * 32];
    __shared__ float s_UwT[20 * 32];
    __shared__ float s_WoT[40 * 16];
    __shared__ float s_outb[32 * 16];  // tanh output head [N pad][O pad]
    __shared__ float s_bzs[32], s_brs[32], s_bws[32], s_bos[16];
    __shared__ float s_cf[4 * 8];      // cfcx
    __shared__ float s_af[14 * 8];     // afcy
    __shared__ float s_rfc[4 * 14];

    const int lane = threadIdx.x & 31;

    // ---- stage & zero-pad inputs ----
    for (int i = lane; i < 1024; i += 32) { s_adj[i] = 0.f; s_h[i] = 0.f; s_x[i] = 0.f; }
    __syncthreads();
    for (int i = lane; i < 18 * 18; i += 32) {
        int r = i / 18, c = i % 18;
        s_adj[r * 32 + c] = adj[i];
    }
    for (int i = lane; i < 18 * 20; i += 32) {
        int r = i / 20, c = i % 20;
        float v = x[i];
        s_h[r * 32 + c] = v;
        s_x[r * 32 + c] = v;
    }
    for (int i = lane; i < 40 * 32; i += 32) {
        int k = i / 32, n = i % 32;
        float wz = 0.f, wr = 0.f, ww = 0.f;
        if (n < 20) { wz = Wz[n * 40 + k]; wr = Wr[n * 40 + k]; ww = Ww[n * 40 + k]; }
        s_WzT[i] = wz; s_WrT[i] = wr; s_WwT[i] = ww;
    }
    for (int i = lane; i < 20 * 32; i += 32) {
        int k = i / 32, n = i % 32;
        float uz = 0.f, ur = 0.f, uw = 0.f;
        if (n < 20) { uz = Uz[n * 20 + k]; ur = Ur[n * 20 + k]; uw = Uw[n * 20 + k]; }
        s_UzT[i] = uz; s_UrT[i] = ur; s_UwT[i] = uw;
    }
    for (int i = lane; i < 40 * 16; i += 32) {
        int k = i / 16, n = i % 16;
        s_WoT[i] = (n < 8) ? Wo[n * 40 + k] : 0.f;
    }
    if (lane < 32) {
        s_bzs[lane] = (lane < 20) ? (bz[lane] + buz[lane]) : 0.f;
        s_brs[lane] = (lane < 20) ? (br[lane] + bur[lane]) : 0.f;
        s_bws[lane] = (lane < 20) ? (bw[lane] + bu[lane])  : 0.f;
    }
    if (lane < 16) s_bos[lane] = (lane < 8) ? bo[lane] : 0.f;
    __syncthreads();

    // ---- GRU recurrence, T = 5 ----
    for (int t = 0; t < 5; ++t) {
        // message aggregation: ain = A.h, aout = A^T.h  (K padded 18->20)
        for (int tm = 0; tm < 2; ++tm)
            for (int tn = 0; tn < 2; ++tn) {
                v8f c = {};
                c = wmma_acc(c, s_adj, 32, tm * 16, s_h, 32, tn * 16, 20, false);
                store_tile(s_ain, 32, tm * 16, tn * 16, c, nullptr, ACT_NONE);
                v8f d = {};
                d = wmma_acc(d, s_adj, 32, tm * 16, s_h, 32, tn * 16, 20, true);
                store_tile(s_aout, 32, tm * 16, tn * 16, d, nullptr, ACT_NONE);
            }
        __syncthreads();

        // z = sigmoid(a.Wz^T + h.Uz^T + bz + buz); r likewise
        for (int tm = 0; tm < 2; ++tm)
            for (int tn = 0; tn < 2; ++tn) {
                v8f c = {};
                c = wmma_acc(c, s_ain,  32, tm * 16, s_WzT,           32, tn * 16, 20, false);
                c = wmma_acc(c, s_aout, 32, tm * 16, s_WzT + 20 * 32, 32, tn * 16, 20, false);
                c = wmma_acc(c, s_h,    32, tm * 16, s_UzT,           32, tn * 16, 20, false);
                store_tile(s_z, 32, tm * 16, tn * 16, c, s_bzs, ACT_SIG);
                v8f d = {};
                d = wmma_acc(d, s_ain,  32, tm * 16, s_WrT,           32, tn * 16, 20, false);
                d = wmma_acc(d, s_aout, 32, tm * 16, s_WrT + 20 * 32, 32, tn * 16, 20, false);
                d = wmma_acc(d, s_h,    32, tm * 16, s_UrT,           32, tn * 16, 20, false);
                store_tile(s_r, 32, tm * 16, tn * 16, d, s_brs, ACT_SIG);
            }
        __syncthreads();

        // r <- r * h (elementwise)
        for (int i = lane; i < 1024; i += 32) s_r[i] *= s_h[i];
        __syncthreads();

        // hg = tanh(a.Ww^T + (r*h).Uw^T + bw + bu)
        for (int tm = 0; tm < 2; ++tm)
            for (int tn = 0; tn < 2; ++tn) {
                v8f c = {};
                c = wmma_acc(c, s_ain,  32, tm * 16, s_WwT,           32, tn * 16, 20, false);
                c = wmma_acc(c, s_aout, 32, tm * 16, s_WwT + 20 * 32, 32, tn * 16, 20, false);
                c = wmma_acc(c, s_r,    32, tm * 16, s_UwT,           32, tn * 16, 20, false);
                store_tile(s_hg, 32, tm * 16, tn * 16, c, s_bws, ACT_TANH);
            }
        __syncthreads();

        // h <- (1-z)*h + z*hg
        for (int i = lane; i < 1024; i += 32) {
            float z = s_z[i];
            s_h[i] = (1.0f - z) * s_h[i] + z * s_hg[i];
        }
        __syncthreads();
    }

    // ---- out = tanh(concat([h, x]) @ Wo^T + bo)  -> [18, 8] ----
    for (int tm = 0; tm < 2; ++tm) {
        v8f c = {};
        c = wmma_acc(c, s_h, 32, tm * 16, s_WoT,           16, 0, 20, false);
        c = wmma_acc(c, s_x, 32, tm * 16, s_WoT + 20 * 16, 16, 0, 20, false);
        store_tile(s_outb, 16, tm * 16, 0, c, s_bos, ACT_TANH);
    }
    __syncthreads();

    // ---- attention head (tiny, VALU) ----
    // cfcx[c][o] = tanh(h[c] . Wx[o] + bx[o])   (4*8 = 32 elems, one per lane)
    for (int i = lane; i < 32; i += 32) {
        int o = i & 7, c = i >> 3;
        float s = bx[o];
        for (int k = 0; k < 20; ++k) s += s_h[c * 32 + k] * Wx[o * 20 + k];
        s_cf[i] = tanhf(s);
    }
    // afcy[a][o] = tanh(h[4+a] . Wy[o] + by[o])  (14*8 = 112 elems)
    for (int i = lane; i < 14 * 8; i += 32) {
        int o = i & 7, a = i >> 3;
        float s = by[o];
        for (int k = 0; k < 20; ++k) s += s_h[(4 + a) * 32 + k] * Wy[o * 20 + k];
        s_af[i] = tanhf(s);
    }
    __syncthreads();

    // rfc[c][a] = sigmoid(sum_o cf*af*w2 + b2) * adj[c][4+a]
    for (int i = lane; i < 4 * 14; i += 32) {
        int a = i % 14, c = i / 14;
        float s = 0.f;
        for (int o = 0; o < 8; ++o) s += s_cf[c * 8 + o] * s_af[a * 8 + o] * w2[o];
        s_rfc[i] = sigf(s + b2[0]) * s_adj[c * 32 + (4 + a)];
    }
    __syncthreads();

    // final [C=4][1+A=15][O=8]: slot 0 = out[c], slots 1..14 = out[4+a] * rfc[c][a]
    for (int i = lane; i < 4 * 15 * 8; i += 32) {
        int o = i & 7;
        int s = (i >> 3) % 15;
        int c = i / (15 * 8);
        float v;
        if (s == 0) v = s_outb[c * 16 + o];
        else        v = s_outb[(4 + (s - 1)) * 16 + o] * s_rfc[c * 14 + (s - 1)];
        out[i] = v;
    }
}

extern "C" void kernel_launch(void* const* d_in, const int* in_sizes, int n_in,
                              void* d_out, int out_size, void* d_ws, size_t ws_size,
                              hipStream_t stream) {
    (void)in_sizes; (void)n_in; (void)out_size; (void)d_ws; (void)ws_size;
    ggnn_kernel<<<1, 32, 0, stream>>>(
        (const float*)d_in[0],  (const float*)d_in[1],
        (const float*)d_in[2],  (const float*)d_in[3],
        (const float*)d_in[4],  (const float*)d_in[5],
        (const float*)d_in[6],  (const float*)d_in[7],
        (const float*)d_in[8],  (const float*)d_in[9],
        (const float*)d_in[10], (const float*)d_in[11],
        (const float*)d_in[12], (const float*)d_in[13],
        (const float*)d_in[14], (const float*)d_in[15],
        (const float*)d_in[16], (const float*)d_in[17],
        (const float*)d_in[18], (const float*)d_in[19],
        (const float*)d_in[20], (const float*)d_in[21],
        (float*)d_out);
}